// Laplacian3D_14946486190476
// MI455X (gfx1250) — compile-verified
//
#include <hip/hip_runtime.h>
#include <cstdint>

// ---------------------------------------------------------------------------
// 7-point Laplacian, B=4, D=H=W=192, fp32, ambient(300K) halo.
// HBM-bound (~226MB traffic, ~10us at 23.3TB/s). Strategy:
//  - Stage (TY+2)-row slabs into LDS via CDNA5 async-to-LDS DMA
//    (global_load_async_to_lds_b128, ASYNCcnt), 4-slab ring buffer.
//  - Prefetch plane z+2 while computing plane z; one s_wait_asynccnt +
//    barrier per plane.
//  - B128 everywhere; output stored non-temporal so the (L2-resident)
//    input halos aren't evicted by write data.
// ---------------------------------------------------------------------------

typedef float v4f __attribute__((ext_vector_type(4)));

namespace {
constexpr int Bn = 4, Dn = 192, Hn = 192, Wn = 192;
constexpr int TY = 16;                     // y-rows computed per block
constexpr int ZC = 24;                     // z-planes marched per block
constexpr int SLAB_ROWS = TY + 2;          // +y halo rows
constexpr int SLAB_F    = SLAB_ROWS * Wn;  // floats per slab (3456)
constexpr int NSLAB     = 4;               // ring: 3 in use + 1 prefetch
constexpr int NTHR      = 256;             // 8 waves (wave32)
constexpr int W4        = Wn / 4;          // float4 per row (48)

constexpr float AMB = 300.0f;
constexpr float WXc = 10000.0f;            // 1/DX^2 (D axis)
constexpr float WYc = 10000.0f;            // 1/DY^2 (H axis)
constexpr float WZc = 10000.0f;            // 1/DZ^2 (W axis)
constexpr float CCc = -2.0f * (WXc + WYc + WZc);
} // namespace

__device__ __forceinline__ void async_load_b128(uint32_t lds_addr, uint64_t gaddr) {
  // GV mode: VDST = LDS byte address, VADDR = 64-bit global address.
  asm volatile("global_load_async_to_lds_b128 %0, %1, off"
               :: "v"(lds_addr), "v"(gaddr)
               : "memory");
}

__device__ __forceinline__ void wait_async0() {
  asm volatile("s_wait_asynccnt 0x0" ::: "memory");
}

__global__ __launch_bounds__(NTHR) void
laplacian3d_async(const float* __restrict__ x, float* __restrict__ out) {
  __shared__ float smem[NSLAB * SLAB_F];

  const int tid = threadIdx.x;
  const int y0  = blockIdx.x * TY;   // 12 tiles
  const int zs  = blockIdx.y * ZC;   // 8 chunks
  const int b   = blockIdx.z;        // 4 batches
  const int ze  = zs + ZC;

  const float* xb   = x   + (size_t)b * Dn * Hn * Wn;
  float*       outb = out + (size_t)b * Dn * Hn * Wn;

  // Stage plane p (rows y0-1 .. y0+TY) into ring slot (p & 3).
  auto load_plane = [&](int p) {
    float* slab = &smem[(p & (NSLAB - 1)) * SLAB_F];
    constexpr int ITEMS = SLAB_ROWS * W4;  // 864 B128 transfers
    if (p < 0 || p >= Dn) {                // z out of range -> ambient slab
      const v4f a = {AMB, AMB, AMB, AMB};
      for (int i = tid; i < ITEMS; i += NTHR)
        *reinterpret_cast<v4f*>(slab + i * 4) = a;
    } else {
      for (int i = tid; i < ITEMS; i += NTHR) {
        const int rr = i / W4;
        const int c4 = i - rr * W4;
        const int gy = y0 - 1 + rr;
        float* ldst = slab + rr * Wn + c4 * 4;
        if (gy < 0 || gy >= Hn) {          // y halo out of range -> ambient
          const v4f a = {AMB, AMB, AMB, AMB};
          *reinterpret_cast<v4f*>(ldst) = a;
        } else {
          const uint64_t ga = (uint64_t)(uintptr_t)
              (xb + ((size_t)p * Hn + gy) * Wn + c4 * 4);
          async_load_b128((uint32_t)(uintptr_t)ldst, ga);
        }
      }
    }
  };

  // Prologue: planes zs-1, zs, zs+1.
  load_plane(zs - 1);
  load_plane(zs);
  load_plane(zs + 1);
  wait_async0();
  __syncthreads();

  // Compute mapping: 256 threads = 16 rows x 16 cols, shifts/ands only.
  const int r  = tid >> 4;           // 0..15  (row within tile)
  const int cx = tid & 15;           // 0..15  (base float4 column)
  const int gy = y0 + r;

  for (int z = zs; z < ze; ++z) {
    // Prefetch plane z+2 (async DMA overlaps the compute below).
    if (z + 2 <= ze) load_plane(z + 2);

    const float* cur = &smem[( z      & (NSLAB - 1)) * SLAB_F];
    const float* zm_ = &smem[((z - 1) & (NSLAB - 1)) * SLAB_F];
    const float* zp_ = &smem[((z + 1) & (NSLAB - 1)) * SLAB_F];

    const float* rc  = cur + (r + 1) * Wn;
    const float* rym = cur + r       * Wn;
    const float* ryp = cur + (r + 2) * Wn;
    const float* rzm = zm_ + (r + 1) * Wn;
    const float* rzp = zp_ + (r + 1) * Wn;
    float* orow = outb + ((size_t)z * Hn + gy) * Wn;

#pragma unroll
    for (int k = 0; k < 3; ++k) {          // 3 * 16 cols = 48 float4 per row
      const int x0 = (cx + (k << 4)) << 2; // float index

      const v4f   c  = *reinterpret_cast<const v4f*>(rc  + x0);
      const float lf = (x0 == 0)      ? AMB : rc[x0 - 1];
      const float rt = (x0 + 4 == Wn) ? AMB : rc[x0 + 4];
      const v4f   ym = *reinterpret_cast<const v4f*>(rym + x0);
      const v4f   yp = *reinterpret_cast<const v4f*>(ryp + x0);
      const v4f   zm = *reinterpret_cast<const v4f*>(rzm + x0);
      const v4f   zp = *reinterpret_cast<const v4f*>(rzp + x0);

      v4f o;
      o.x = CCc * c.x + WXc * (zm.x + zp.x) + WYc * (ym.x + yp.x) + WZc * (lf  + c.y);
      o.y = CCc * c.y + WXc * (zm.y + zp.y) + WYc * (ym.y + yp.y) + WZc * (c.x + c.z);
      o.z = CCc * c.z + WXc * (zm.z + zp.z) + WYc * (ym.z + yp.z) + WZc * (c.y + c.w);
      o.w = CCc * c.w + WXc * (zm.w + zp.w) + WYc * (ym.w + yp.w) + WZc * (c.z + rt);

      // Streaming store: output is write-once, keep L2 for input halos.
      __builtin_nontemporal_store(o, reinterpret_cast<v4f*>(orow + x0));
    }

    wait_async0();      // prefetch of z+2 landed in LDS
    __syncthreads();    // visible to all waves; safe to reuse ring slot
  }
}

extern "C" void kernel_launch(void* const* d_in, const int* in_sizes, int n_in,
                              void* d_out, int out_size, void* d_ws, size_t ws_size,
                              hipStream_t stream) {
  (void)in_sizes; (void)n_in; (void)d_ws; (void)ws_size; (void)out_size;
  const float* x = (const float*)d_in[0];
  float* out = (float*)d_out;

  dim3 grid(Hn / TY, Dn / ZC, Bn);   // 12 x 8 x 4 = 384 blocks
  dim3 block(NTHR);
  laplacian3d_async<<<grid, block, 0, stream>>>(x, out);
}